// IoULoss_36197984371322
// MI455X (gfx1250) — compile-verified
//
#include <hip/hip_runtime.h>
#include <hip/hip_bf16.h>

typedef float v2f __attribute__((ext_vector_type(2)));
typedef float v8f __attribute__((ext_vector_type(8)));

// ---------------------------------------------------------------------------
// Wave32 cross-lane sum via V_WMMA_F32_16X16X4_F32.
// A (16x4 f32): lanes 0-15 -> {K0=VGPR0, K1=VGPR1}, lanes 16-31 -> {K2,K3}.
// We place each lane's partial in A element 0 and zero in element 1.
// With B = all-ones (layout-invariant) and C = 0:
//   D[i][j] = rowsum_i = local[lane i] + local[lane i+16]   (same for all j)
// C/D layout: lane j<16 holds M=0..7 in VGPR0..7 (N=j); lane j>=16 holds
// M=8..15. So summing a lane's 8 D registers gives:
//   lanes  0-15 : sum of locals over lanes 0-7 and 16-23
//   lanes 16-31 : sum of locals over lanes 8-15 and 24-31
// Caller combines the two halves (via LDS). EXEC is all-1s at the call sites.
// ---------------------------------------------------------------------------
__device__ __forceinline__ float wave_halfsum_wmma(float v) {
    v2f a; a[0] = v;    a[1] = 0.0f;
    v2f b; b[0] = 1.0f; b[1] = 1.0f;
    v8f c = {};
    v8f d = __builtin_amdgcn_wmma_f32_16x16x4_f32(
        /*neg_a=*/false, a, /*neg_b=*/false, b,
        /*c_mod=*/(short)0, c, /*reuse_a=*/false, /*reuse_b=*/false);
    return ((d[0] + d[1]) + (d[2] + d[3])) + ((d[4] + d[5]) + (d[6] + d[7]));
}

// Block-wide deterministic reduction: per-wave WMMA halves -> LDS -> lane 0.
// blockDim.x must be 256 (8 waves).
__device__ __forceinline__ float block_reduce_256(float v, float* lds16) {
    const int lane = threadIdx.x & 31;
    const int wid  = threadIdx.x >> 5;
    float h = wave_halfsum_wmma(v);          // all 32 lanes execute (EXEC full)
    if ((lane & 15) == 0)                    // lanes 0 and 16 hold distinct halves
        lds16[wid * 2 + (lane >> 4)] = h;    // identical-value WAW within half is benign
    __syncthreads();
    float s = 0.0f;
    if (threadIdx.x == 0) {
        #pragma unroll
        for (int i = 0; i < 16; ++i) s += lds16[i];
    }
    return s;                                // valid in thread 0 only
}

__device__ __forceinline__ float smooth_l1(float x, float y) {
    float d = fabsf(x - y);
    return (d < 1.0f) ? 0.5f * d * d : d - 0.5f;
}

// ---------------------------------------------------------------------------
// Stage 1: streaming partial reduction. One float4 per box per tensor
// (global_load_b128, fully coalesced). Each box contributes
//   iou + 0.25 * (sum of 4 smooth-L1 terms)
// so the final answer is total / n_boxes.
// ---------------------------------------------------------------------------
__global__ __launch_bounds__(256) void iou_partial_kernel(
    const float4* __restrict__ outp, const float4* __restrict__ tgt,
    float* __restrict__ partials, int n_boxes) {
    __shared__ float lds16[16];

    float local = 0.0f;
    const int stride = gridDim.x * blockDim.x;
    for (int i = blockIdx.x * blockDim.x + threadIdx.x; i < n_boxes; i += stride) {
        const float4 a = outp[i];
        const float4 b = tgt[i];

        // IoU (diagonal)
        const float ltx = fmaxf(a.x, b.x), lty = fmaxf(a.y, b.y);
        const float rbx = fminf(a.z, b.z), rby = fminf(a.w, b.w);
        const float w   = fmaxf(rbx - ltx, 0.0f);
        const float h   = fmaxf(rby - lty, 0.0f);
        const float inter  = w * h;
        const float area_a = (a.z - a.x) * (a.w - a.y);
        const float area_b = (b.z - b.x) * (b.w - b.y);
        const float iou    = inter / (area_a + area_b - inter);

        // smooth L1 over the 4 coords, pre-scaled by 1/4
        const float l1 = smooth_l1(a.x, b.x) + smooth_l1(a.y, b.y) +
                         smooth_l1(a.z, b.z) + smooth_l1(a.w, b.w);

        local += iou + 0.25f * l1;
    }

    const float s = block_reduce_256(local, lds16);
    if (threadIdx.x == 0) partials[blockIdx.x] = s;
}

// ---------------------------------------------------------------------------
// Stage 2: single block folds the per-block partials and scales by 1/n_boxes.
// ---------------------------------------------------------------------------
__global__ __launch_bounds__(256) void iou_final_kernel(
    const float* __restrict__ partials, float* __restrict__ out,
    int n_partials, float inv_n) {
    __shared__ float lds16[16];

    float local = 0.0f;
    for (int i = threadIdx.x; i < n_partials; i += blockDim.x)
        local += partials[i];

    const float s = block_reduce_256(local, lds16);
    if (threadIdx.x == 0) out[0] = s * inv_n;
}

extern "C" void kernel_launch(void* const* d_in, const int* in_sizes, int n_in,
                              void* d_out, int out_size, void* d_ws, size_t ws_size,
                              hipStream_t stream) {
    const float4* outp = (const float4*)d_in[0];
    const float4* tgt  = (const float4*)d_in[1];
    float* partials    = (float*)d_ws;
    float* out         = (float*)d_out;

    const int n_boxes = in_sizes[0] / 4;          // B*C = 5,242,880

    int blocks = 2048;                            // ~524k threads: saturate 23.3 TB/s HBM
    const int max_ws_blocks = (int)(ws_size / sizeof(float));
    if (blocks > max_ws_blocks) blocks = max_ws_blocks;
    const int max_useful = (n_boxes + 255) / 256;
    if (blocks > max_useful) blocks = max_useful;
    if (blocks < 1) blocks = 1;

    const float inv_n = (float)(1.0 / (double)n_boxes);

    iou_partial_kernel<<<blocks, 256, 0, stream>>>(outp, tgt, partials, n_boxes);
    iou_final_kernel<<<1, 256, 0, stream>>>(partials, out, blocks, inv_n);
}